// GraphAddPooling_39539468927441
// MI455X (gfx1250) — compile-verified
//
#include <hip/hip_runtime.h>
#include <hip/hip_bf16.h>

typedef __attribute__((ext_vector_type(2))) float v2f;
typedef __attribute__((ext_vector_type(8))) float v8f;

#define NUM_SEG 512
#define SEG_PER_GROUP 16
#define COLS 256
#define P_ROW 16  // row partitions per segment-group

__device__ __forceinline__ int lower_bound_i64(const long long* __restrict__ b, int n, long long key) {
    int lo = 0, hi = n;
    while (lo < hi) {
        int mid = (lo + hi) >> 1;
        if (b[mid] < key) lo = mid + 1; else hi = mid;
    }
    return lo;
}

__global__ void zero_out_kernel(float* __restrict__ out, int n) {
    int i = blockIdx.x * blockDim.x + threadIdx.x;
    if (i < n) out[i] = 0.0f;
}

// One block = 512 threads = 16 waves; wave w owns columns [16w, 16w+16).
// blockIdx.x = segment group (16 consecutive segments), blockIdx.y = row partition.
// Each iteration consumes 16 rows via 4x V_WMMA_F32_16X16X4_F32 with a one-hot
// selector as the A matrix (exact f32 segment scatter-add).
__global__ void __launch_bounds__(512)
GraphAddPooling_segsum_wmma_kernel(const float* __restrict__ x,
                                   const long long* __restrict__ batch,
                                   float* __restrict__ out,
                                   int nrows) {
    const int lane = threadIdx.x & 31;
    const int wave = threadIdx.x >> 5;          // 0..15 -> column tile
    const int seg_base = blockIdx.x * SEG_PER_GROUP;
    const int part = blockIdx.y;
    const int col_base = wave * 16;

    // Row range [s,e) of this segment group (batch is sorted). Uniform per block.
    const int s = lower_bound_i64(batch, nrows, (long long)seg_base);
    const int e = lower_bound_i64(batch, nrows, (long long)(seg_base + SEG_PER_GROUP));
    const int total = e - s;
    const int chunk = (total + P_ROW - 1) / P_ROW;
    const int pr0 = s + part * chunk;
    const int pr1 = min(e, pr0 + chunk);

    const int n = lane & 15;                    // column within tile == segment-m for A
    const int base_k = (lane < 16) ? 0 : 2;     // half-wave K split (A and B layouts)
    const int nclamp = nrows - 1;
    const int* __restrict__ b32 = (const int*)batch;  // low dword of int64 (values < 512)

    v8f acc = {};

    for (int r = pr0; r < pr1; r += 16) {
        v2f A[4], B[4];

        // Gather 16 rows worth of selectors + data; branchless so loads clause up.
        #pragma unroll
        for (int u = 0; u < 4; ++u) {
            const int row0 = r + 4 * u + base_k;
            const int row1 = row0 + 1;
            const int row0c = min(row0, nclamp);
            const int row1c = min(row1, nclamp);

            const int bv0 = b32[row0c << 1];
            const int bv1 = b32[row1c << 1];
            const int ok0 = (int)(row0 < pr1) & (int)(bv0 == seg_base + n);
            const int ok1 = (int)(row1 < pr1) & (int)(bv1 == seg_base + n);
            A[u].x = ok0 ? 1.0f : 0.0f;
            A[u].y = ok1 ? 1.0f : 0.0f;

            // x is a single-use 205MB stream: non-temporal, keep L2 for out/batch.
            B[u].x = __builtin_nontemporal_load(&x[(size_t)row0c * COLS + col_base + n]);
            B[u].y = __builtin_nontemporal_load(&x[(size_t)row1c * COLS + col_base + n]);
        }

        // Prefetch the row stream ahead (global_prefetch_b8).
        __builtin_prefetch(&x[(size_t)min(r + 256, nclamp) * COLS + col_base + n], 0, 1);

        #pragma unroll
        for (int u = 0; u < 4; ++u) {
            // D = A(16x4 one-hot) * B(4x16 of x) + C
            acc = __builtin_amdgcn_wmma_f32_16x16x4_f32(
                /*neg_a=*/false, A[u], /*neg_b=*/false, B[u],
                /*c_mod=*/(short)0, acc, /*reuse_a=*/false, /*reuse_b=*/false);
        }
    }

    // Commit: VGPR j holds M = j + (lane<16 ? 0 : 8), N = lane&15.
    const int m_off = (lane < 16) ? 0 : 8;
    #pragma unroll
    for (int j = 0; j < 8; ++j) {
        const int seg = seg_base + j + m_off;
        atomicAdd(&out[(size_t)seg * COLS + col_base + n], acc[j]);
    }
}

extern "C" void kernel_launch(void* const* d_in, const int* in_sizes, int n_in,
                              void* d_out, int out_size, void* d_ws, size_t ws_size,
                              hipStream_t stream) {
    const float* x = (const float*)d_in[0];
    const long long* batch = (const long long*)d_in[1];
    float* out = (float*)d_out;
    const int nrows = in_sizes[1];

    // Zero the output (harness poisons it), then accumulate.
    zero_out_kernel<<<(out_size + 255) / 256, 256, 0, stream>>>(out, out_size);

    dim3 grid(NUM_SEG / SEG_PER_GROUP, P_ROW, 1);  // 32 x 16
    GraphAddPooling_segsum_wmma_kernel<<<grid, 512, 0, stream>>>(x, batch, out, nrows);
}